// LayerNormDenseGeneral_3092376453257
// MI455X (gfx1250) — compile-verified
//
#include <hip/hip_runtime.h>
#include <stdint.h>

// ---------------------------------------------------------------------------
// x[S,B,H] -> LayerNorm -> y ; z = y @ W[H,F]
//   M = S*B = 8192, K = H = 2048, N = F = 8192, fp32.
// d_out: z (M*N floats) then y (M*K floats).
// ---------------------------------------------------------------------------
#define M_DIM 8192
#define K_DIM 2048
#define N_DIM 8192
#define LN_EPS 1e-6f

typedef __attribute__((ext_vector_type(2))) float v2f;
typedef __attribute__((ext_vector_type(8))) float v8f;
typedef __attribute__((ext_vector_type(4))) unsigned int v4u;
typedef __attribute__((ext_vector_type(8))) int v8i;
typedef __attribute__((ext_vector_type(4))) int v4i;

// ---------------------------------------------------------------------------
// Kernel 1: LayerNorm rows (bandwidth bound: 64 MiB in + 64 MiB out)
// ---------------------------------------------------------------------------
__global__ void __launch_bounds__(256)
ln_rows_kernel(const float* __restrict__ x,
               const float* __restrict__ scale,
               const float* __restrict__ bias,
               float* __restrict__ y)
{
    __shared__ float s_sum[256];
    __shared__ float s_sq[256];

    const int row = blockIdx.x;
    const int tid = threadIdx.x;
    const float* xr = x + (size_t)row * K_DIM;
    float*       yr = y + (size_t)row * K_DIM;

    const int h0 = tid * 4;
    const int h1 = 1024 + tid * 4;
    float4 a = *(const float4*)(xr + h0);
    float4 b = *(const float4*)(xr + h1);

    float sum = a.x + a.y + a.z + a.w + b.x + b.y + b.z + b.w;
    float sq  = a.x*a.x + a.y*a.y + a.z*a.z + a.w*a.w
              + b.x*b.x + b.y*b.y + b.z*b.z + b.w*b.w;

    s_sum[tid] = sum;
    s_sq[tid]  = sq;
    __syncthreads();
    for (int off = 128; off > 0; off >>= 1) {
        if (tid < off) {
            s_sum[tid] += s_sum[tid + off];
            s_sq[tid]  += s_sq[tid + off];
        }
        __syncthreads();
    }
    const float inv_n = 1.0f / (float)K_DIM;
    const float mu    = s_sum[0] * inv_n;
    const float var   = s_sq[0] * inv_n - mu * mu;
    const float rstd  = rsqrtf(var + LN_EPS);

    float4 g0 = *(const float4*)(scale + h0);
    float4 g1 = *(const float4*)(scale + h1);
    float4 c0 = *(const float4*)(bias + h0);
    float4 c1 = *(const float4*)(bias + h1);

    float4 o0, o1;
    o0.x = (a.x - mu) * rstd * g0.x + c0.x;
    o0.y = (a.y - mu) * rstd * g0.y + c0.y;
    o0.z = (a.z - mu) * rstd * g0.z + c0.z;
    o0.w = (a.w - mu) * rstd * g0.w + c0.w;
    o1.x = (b.x - mu) * rstd * g1.x + c1.x;
    o1.y = (b.y - mu) * rstd * g1.y + c1.y;
    o1.z = (b.z - mu) * rstd * g1.z + c1.z;
    o1.w = (b.w - mu) * rstd * g1.w + c1.w;

    *(float4*)(yr + h0) = o0;
    *(float4*)(yr + h1) = o1;
}

// ---------------------------------------------------------------------------
// TDM: issue a 2D tensor_load_to_lds (one wave issues; EXEC ignored).
// Descriptor per CDNA5 ISA 8.3/8.4:
//   group0: count=1 | lds_addr | global_addr[56:0] | type=2
//   group1: data_size=4B, pad_enable, pad_interval/amount, tensor/tile dims,
//           tensor_dim0_stride (elements)
// pad_interval_code: LDS pad inserted every 2^(code+1) DWORDs
// pad_amount_code:   pad of (code+1) DWORDs
// ---------------------------------------------------------------------------
__device__ __forceinline__ void tdm_load_2d(uint32_t lds_byte_addr,
                                            const float* gptr,
                                            uint32_t tensor_d0, uint32_t tensor_d1,
                                            uint32_t tile_d0,   uint32_t tile_d1,
                                            uint32_t stride0_elems,
                                            uint32_t pad_interval_code,
                                            uint32_t pad_amount_code)
{
    const uint64_t ga = (uint64_t)(uintptr_t)gptr;

    v4u g0;
    g0[0] = 1u;                                              // count=1 (valid), user mode
    g0[1] = lds_byte_addr;                                   // LDS dest (bytes)
    g0[2] = (uint32_t)ga;                                    // global_addr[31:0]
    g0[3] = (uint32_t)((ga >> 32) & 0x01FFFFFFu) | (2u << 30); // addr[56:32], type=2

    unsigned w0 = (2u << 16)                                 // data_size = 4 bytes
                | (1u << 20)                                 // pad_enable
                | (pad_interval_code << 22)
                | (pad_amount_code << 25);
    unsigned w1 = (tensor_d0 & 0xFFFFu) << 16;               // tensor_dim0[15:0]
    unsigned w2 = (tensor_d0 >> 16) | ((tensor_d1 & 0xFFFFu) << 16);
    unsigned w3 = (tensor_d1 >> 16) | (tile_d0 << 16);       // tile_dim0
    unsigned w4 = tile_d1;                                   // tile_dim1 (tile_dim2=0)
    unsigned w5 = stride0_elems;                             // tensor_dim0_stride[31:0]
    unsigned w6 = 0;                                         // stride0[47:32]=0, dim1_stride lo=0
    unsigned w7 = 0;                                         // dim1_stride hi=0

    v8i g1;
    g1[0] = (int)w0; g1[1] = (int)w1; g1[2] = (int)w2; g1[3] = (int)w3;
    g1[4] = (int)w4; g1[5] = (int)w5; g1[6] = (int)w6; g1[7] = (int)w7;

    v4i gz4 = {0, 0, 0, 0};
    v8i gz8 = {0, 0, 0, 0, 0, 0, 0, 0};

    // 6-arg form (clang-23 / therock headers): (g0, g1, g2, g3, g4, cpol)
    __builtin_amdgcn_tensor_load_to_lds(g0, g1, gz4, gz4, gz8, /*cpol=*/0);
}

// ---------------------------------------------------------------------------
// Kernel 2: z = y @ W via V_WMMA_F32_16X16X4_F32, TDM double-buffered LDS.
//
// Block tile 128x128, BK=32; 8 waves (2Mx4N), each wave: 64x32 = 4x2 frags.
// LDS (dynamic, 70656 B): A[2][128][36], B[2][32][132] fp32 (TDM-padded rows).
// f32 WMMA layouts (ISA 7.12.2):
//   A 16x4 : lane<16 row=l {K0,K1}; lane>=16 row=l-16 {K2,K3}
//   B 4x16 : lane<16 col=l {K0,K1}; lane>=16 col=l-16 {K2,K3}
//   C 16x16: VGPR j -> row j + 8*(lane>=16), col lane&15
// ---------------------------------------------------------------------------
#define BM 128
#define BN 128
#define BK 32
#define A_STRIDE 36
#define B_STRIDE 132
#define A_BYTES  (BM * A_STRIDE * 4)   // 18432
#define B_BYTES  (BK * B_STRIDE * 4)   // 16896
#define LDS_TOTAL (2 * (A_BYTES + B_BYTES))  // 70656

__global__ void __launch_bounds__(256)
gemm_f32_wmma_tdm_kernel(const float* __restrict__ A,   // y [M,K]
                         const float* __restrict__ B,   // W [K,N]
                         float* __restrict__ C)         // z [M,N]
{
    extern __shared__ char smem[];
    const uint32_t lds_base = (uint32_t)(uintptr_t)smem;

    const int tid  = threadIdx.x;
    const int lane = tid & 31;
    const int wave = tid >> 5;        // 0..7
    const int wm   = wave >> 2;       // 0..1 -> 64-row slice
    const int wn   = wave & 3;        // 0..3 -> 32-col slice

    const int rowBase = blockIdx.y * BM;
    const int colBase = blockIdx.x * BN;

    const int l15   = lane & 15;
    const int khalf = (lane >> 4) * 2;   // 0 or 2

    v8f acc[4][2];
    #pragma unroll
    for (int mi = 0; mi < 4; ++mi)
        #pragma unroll
        for (int ni = 0; ni < 2; ++ni)
            acc[mi][ni] = (v8f){0.f,0.f,0.f,0.f,0.f,0.f,0.f,0.f};

    // ---- prologue: TDM-load tiles for k0 = 0 into buffer 0 ----
    if (wave == 0) {
        // A tile: 128 rows x 32 cols, row stride K; LDS stride 36 (pad 32DW/+4DW)
        tdm_load_2d(lds_base, A + (size_t)rowBase * K_DIM,
                    K_DIM, M_DIM, BK, BM, K_DIM, /*intv 32DW*/4, /*pad 4DW*/3);
        // B tile: 32 rows x 128 cols, row stride N; LDS stride 132 (pad 128DW/+4DW)
        tdm_load_2d(lds_base + 2 * A_BYTES, B + colBase,
                    N_DIM, K_DIM, BN, BK, N_DIM, /*intv 128DW*/6, /*pad 4DW*/3);
    }

    int cur = 0;
    for (int k0 = 0; k0 < K_DIM; k0 += BK) {
        const int nxt = cur ^ 1;
        if (wave == 0) {
            if (k0 + BK < K_DIM) {
                const int kn = k0 + BK;
                tdm_load_2d(lds_base + (uint32_t)nxt * A_BYTES,
                            A + (size_t)rowBase * K_DIM + kn,
                            K_DIM, M_DIM, BK, BM, K_DIM, 4, 3);
                tdm_load_2d(lds_base + 2 * A_BYTES + (uint32_t)nxt * B_BYTES,
                            B + (size_t)kn * N_DIM + colBase,
                            N_DIM, K_DIM, BN, BK, N_DIM, 6, 3);
                __builtin_amdgcn_s_wait_tensorcnt(2);   // current pair done
            } else {
                __builtin_amdgcn_s_wait_tensorcnt(0);   // last pair done
            }
        }
        __syncthreads();   // LDS tiles visible to all waves

        const float* Alds = (const float*)(smem + (size_t)cur * A_BYTES);
        const float* Blds = (const float*)(smem + 2 * A_BYTES + (size_t)cur * B_BYTES);

        #pragma unroll
        for (int kk = 0; kk < BK; kk += 4) {
            v2f afrag[4];
            #pragma unroll
            for (int mi = 0; mi < 4; ++mi) {
                const int row = wm * 64 + mi * 16 + l15;
                afrag[mi] = *(const v2f*)(&Alds[row * A_STRIDE + kk + khalf]);
            }
            v2f bfrag[2];
            #pragma unroll
            for (int ni = 0; ni < 2; ++ni) {
                const int col  = wn * 32 + ni * 16 + l15;
                const int krow = kk + khalf;
                v2f t;
                t.x = Blds[krow * B_STRIDE + col];
                t.y = Blds[(krow + 1) * B_STRIDE + col];
                bfrag[ni] = t;
            }
            #pragma unroll
            for (int mi = 0; mi < 4; ++mi)
                #pragma unroll
                for (int ni = 0; ni < 2; ++ni)
                    acc[mi][ni] = __builtin_amdgcn_wmma_f32_16x16x4_f32(
                        false, afrag[mi], false, bfrag[ni],
                        (short)0, acc[mi][ni], false, false);
        }
        __syncthreads();   // reads done before buffer is overwritten
        cur = nxt;
    }

    // ---- epilogue: scatter C fragments ----
    const int rhalf = (lane >> 4) * 8;
    #pragma unroll
    for (int mi = 0; mi < 4; ++mi) {
        #pragma unroll
        for (int ni = 0; ni < 2; ++ni) {
            const int col = colBase + wn * 32 + ni * 16 + l15;
            #pragma unroll
            for (int j = 0; j < 8; ++j) {
                const int row = rowBase + wm * 64 + mi * 16 + rhalf + j;
                C[(size_t)row * N_DIM + col] = acc[mi][ni][j];
            }
        }
    }
}

// ---------------------------------------------------------------------------
// Entry point
// ---------------------------------------------------------------------------
extern "C" void kernel_launch(void* const* d_in, const int* in_sizes, int n_in,
                              void* d_out, int out_size, void* d_ws, size_t ws_size,
                              hipStream_t stream) {
    (void)in_sizes; (void)n_in; (void)out_size; (void)d_ws; (void)ws_size;

    const float* x     = (const float*)d_in[0];   // [S,B,H] = [M,K]
    const float* scale = (const float*)d_in[1];   // [H]
    const float* bias  = (const float*)d_in[2];   // [H]
    const float* W     = (const float*)d_in[3];   // [H,F] = [K,N]

    float* z = (float*)d_out;                     // [M,N]
    float* y = z + (size_t)M_DIM * N_DIM;         // [M,K]

    ln_rows_kernel<<<M_DIM, 256, 0, stream>>>(x, scale, bias, y);

    dim3 grid(N_DIM / BN, M_DIM / BM);
    gemm_f32_wmma_tdm_kernel<<<grid, 256, LDS_TOTAL, stream>>>(y, W, z);
}